// Attn2Quad_87411174408375
// MI455X (gfx1250) — compile-verified
//
#include <hip/hip_runtime.h>

// MI455X / gfx1250, wave32. Fused (QK^T -> epilogue -> scores store -> S*V)
// attention-without-softmax kernel using V_WMMA_F32_16X16X4_F32.
//
// Roofline: output = out (25 MB) + scores (805 MB) fp32. The 805 MB scores
// store stream dominates (~36 us at 23.3 TB/s). So: fuse both GEMMs (never
// re-read scores), stream scores with TH=NT b128 stores (don't thrash the
// 192 MB L2 that serves the K/V re-reads), keep all math in fp32 WMMA.

typedef __attribute__((ext_vector_type(2))) float v2f;
typedef __attribute__((ext_vector_type(4))) float v4f;
typedef __attribute__((ext_vector_type(8))) float v8f;

#define B_ 4
#define H_ 12
#define S_ 2048
#define D_ 64

static constexpr float C_CONST      = 4.0f;
static constexpr float STATIC_SCALE = 0.01f;

// LDS row stride (floats) for the per-wave 16x16 score tile.
// 20 floats = 80 B: keeps 16-byte alignment for b128 re-reads, and
// 5n mod 16 is a permutation -> transpose reads stay bank-conflict-free.
#define SROW 20

__global__ __launch_bounds__(256) void attn2quad_kernel(
    const float* __restrict__ Q, const float* __restrict__ K,
    const float* __restrict__ V, const float* __restrict__ MASK,
    const float* __restrict__ TAU, float* __restrict__ OUT,
    float* __restrict__ SC)
{
    // one wave = 16 queries; 8 waves per block = 128 queries
    __shared__ float lds_s[8 * 16 * SROW];

    const int tid   = threadIdx.x;
    const int wave  = tid >> 5;
    const int lane  = tid & 31;
    const int n16   = lane & 15;   // column / row-in-16 index
    const int uh    = lane >> 4;   // half-wave select
    const int r2    = lane >> 1;   // 0..15: row for wide scores store
    const int c8    = (lane & 1) * 8;  // 0/8: col group for wide scores store

    const int bh   = blockIdx.x >> 4;   // 0..47  (b*H + h)
    const int qblk = blockIdx.x & 15;   // 0..15
    const int b    = bh / H_;
    const int h    = bh - b * H_;
    const int q0   = qblk * 128 + wave * 16;

    const float inv_tau = STATIC_SCALE / TAU[h];

    const float* Qb = Q   + (size_t)bh * S_ * D_;
    const float* Kb = K   + (size_t)bh * S_ * D_;
    const float* Vb = V   + (size_t)bh * S_ * D_;
    float*       Ob = OUT + (size_t)bh * S_ * D_;
    float*       Sb = SC  + (size_t)bh * S_ * S_;
    const float* Mb = MASK + (size_t)b * S_;

    // ---- load Q as A-fragments for all 16 K-steps (d = 4*kk + 2*uh + {0,1})
    // A 16x4 f32 layout: lane L -> M = L%16, VGPR0/1 hold K = 2*(L/16)+{0,1}
    v2f qa[16];
#pragma unroll
    for (int kk = 0; kk < 16; ++kk) {
        const float* p = Qb + (size_t)(q0 + n16) * D_ + 4 * kk + 2 * uh;
        qa[kk] = *(const v2f*)p;
    }

    // O accumulators: 16 queries x 64 d  (4 N-tiles of 16)
    v8f o[4];
#pragma unroll
    for (int nt = 0; nt < 4; ++nt) o[nt] = (v8f){};

    float* my_lds = lds_s + wave * (16 * SROW);

    for (int kt = 0; kt < S_ / 16; ++kt) {
        const int key0 = kt * 16;

        // ---- B-fragments of K^T (B[d][key] = K[key][d]); same per-lane
        // pattern as A: lane L -> N(key) = L%16, K(d) = 2*(L/16)+{0,1}
        v2f kb[16];
#pragma unroll
        for (int kk = 0; kk < 16; ++kk) {
            const float* p = Kb + (size_t)(key0 + n16) * D_ + 4 * kk + 2 * uh;
            kb[kk] = *(const v2f*)p;
        }

        // ---- GEMM 1: s(16x16) = Q(16x64) * K^T(64x16), fp32 WMMA chain
        v8f s = (v8f){};
#pragma unroll
        for (int kk = 0; kk < 16; ++kk) {
            s = __builtin_amdgcn_wmma_f32_16x16x4_f32(
                    /*neg_a=*/false, qa[kk], /*neg_b=*/false, kb[kk],
                    /*c_mod=*/(short)0, s, /*reuse_a=*/false, /*reuse_b=*/false);
        }

        // ---- epilogue: (s + c)^2 * scale / tau * mask[key]
        // C/D layout: lane L holds N = L%16, rows M = r + 8*(L/16)
        const float f = inv_tau * Mb[key0 + n16];
#pragma unroll
        for (int r = 0; r < 8; ++r) {
            float e = s[r] + C_CONST;
            s[r] = e * e * f;
        }

        // ---- stage tile into LDS (source for both the transposed
        // A-fragments and the coalesced wide scores store)
#pragma unroll
        for (int r = 0; r < 8; ++r) {
            const int m = r + 8 * uh;
            my_lds[m * SROW + n16] = s[r];
        }

        // ---- stream scores tile to global: lane L -> row L/2, cols
        // (L&1)*8..+7; two b128 non-temporal stores per lane (512 B/instr
        // wave-wide; NT keeps the 805 MB stream out of L2).
        {
            const float* src = &my_lds[r2 * SROW + c8];
            float* dst = &Sb[(size_t)(q0 + r2) * S_ + key0 + c8];
            v4f w0 = *(const v4f*)(src);
            v4f w1 = *(const v4f*)(src + 4);
            __builtin_nontemporal_store(w0, (v4f*)(dst));
            __builtin_nontemporal_store(w1, (v4f*)(dst + 4));
        }

        // ---- GEMM 2: O(16x64) += s(16x16) * V(16x64)
        // A-fragments of s come from LDS (cross-lane transpose of C layout)
#pragma unroll
        for (int j = 0; j < 4; ++j) {
            v2f a2 = *(const v2f*)&my_lds[n16 * SROW + 4 * j + 2 * uh];
#pragma unroll
            for (int nt = 0; nt < 4; ++nt) {
                // B-fragment of V: lane L -> N(d) = 16*nt + L%16,
                // K(key) = key0 + 4*j + 2*(L/16) + {0,1}
                v2f b2;
                b2.x = Vb[(size_t)(key0 + 4 * j + 2 * uh    ) * D_ + 16 * nt + n16];
                b2.y = Vb[(size_t)(key0 + 4 * j + 2 * uh + 1) * D_ + 16 * nt + n16];
                o[nt] = __builtin_amdgcn_wmma_f32_16x16x4_f32(
                            false, a2, false, b2, (short)0, o[nt], false, false);
            }
        }
    }

    // ---- store O tile (small: 25 MB total, RT policy is fine)
#pragma unroll
    for (int nt = 0; nt < 4; ++nt) {
#pragma unroll
        for (int r = 0; r < 8; ++r) {
            const int m = r + 8 * uh;
            Ob[(size_t)(q0 + m) * D_ + 16 * nt + n16] = o[nt][r];
        }
    }
}

extern "C" void kernel_launch(void* const* d_in, const int* in_sizes, int n_in,
                              void* d_out, int out_size, void* d_ws, size_t ws_size,
                              hipStream_t stream) {
    const float* q    = (const float*)d_in[0];
    const float* k    = (const float*)d_in[1];
    const float* v    = (const float*)d_in[2];
    const float* mask = (const float*)d_in[3];
    const float* tau  = (const float*)d_in[4];

    float* out    = (float*)d_out;                         // [B,H,S,D]
    float* scores = out + (size_t)B_ * H_ * S_ * D_;       // [B,H,S,S]

    dim3 grid(B_ * H_ * (S_ / 128));   // 768 blocks
    dim3 block(256);                   // 8 waves of 32
    attn2quad_kernel<<<grid, block, 0, stream>>>(q, k, v, mask, tau, out, scores);
}